// BuildCostVolume_85581518340313
// MI455X (gfx1250) — compile-verified
//
#include <hip/hip_runtime.h>
#include <hip/hip_bf16.h>
#include <math.h>

typedef __attribute__((ext_vector_type(2))) float v2f;
typedef __attribute__((ext_vector_type(8))) float v8f;

#define HH 64
#define WW 128
#define DVv 24
#define DIN 64
#define DINNER 64
#define DSTATE 8
#define GG 8
#define PTOT (DVv*HH*WW)   // 196608

__device__ __forceinline__ float softplusf_(float z) {
  return z > 20.f ? z : log1pf(expf(z));
}
__device__ __forceinline__ float sigmoidf_(float z) {
  return 1.f / (1.f + expf(-z));
}

// ---------------------------------------------------------------------------
// K1: fused projection GEMM.  Rows = P positions (d,h,w), K = 64 interleaved
// channels (even = left, odd = right shifted by d, masked w>=d), N = 144
// (= W_in 64 | W_dt 64 | W_B 8 | W_C 8).  One wave per 16-row tile; A
// fragments for the whole K sweep live in registers and are reused across the
// 9 N-tiles; B fragments are double-buffered in registers so the 16
// v_wmma_f32_16x16x4_f32 per tile issue back-to-back while the next tile's
// weights are in flight.
// ---------------------------------------------------------------------------
__global__ void __launch_bounds__(128)
proj_gemm_kernel(const float* __restrict__ featL, const float* __restrict__ featR,
                 const float* __restrict__ W_in, const float* __restrict__ W_dt,
                 const float* __restrict__ b_dt, const float* __restrict__ W_B,
                 const float* __restrict__ W_C,
                 float* __restrict__ U, float* __restrict__ DT,
                 float* __restrict__ Bc, float* __restrict__ Cc)
{
  const int lane = threadIdx.x & 31;
  const int wv   = threadIdx.x >> 5;
  const int tile = blockIdx.x * 4 + wv;       // 16-row tile index
  const int p0   = tile * 16;
  const int m    = lane & 15;                 // A row within tile
  const int kg   = lane >> 4;                 // lane half -> K sub-block

  const int p = p0 + m;
  const int w = p & (WW - 1);
  const int h = (p >> 7) & (HH - 1);
  const int d = p >> 13;
  const bool valid = (w >= d);
  const int  wr    = valid ? (w - d) : 0;

  // A fragments: a.x = x[p][4ks+2kg] (left ch), a.y = x[p][4ks+2kg+1] (right)
  v2f afrag[16];
  const float* pL = featL + h * WW + w;
  const float* pR = featR + h * WW + wr;
#pragma unroll
  for (int ks = 0; ks < 16; ++ks) {
    const int kp = 2 * ks + kg;               // source feature channel 0..31
    afrag[ks].x = valid ? pL[kp * (HH * WW)] : 0.f;
    afrag[ks].y = valid ? pR[kp * (HH * WW)] : 0.f;
  }

  const int n = lane & 15;                    // B/D column within tile

  auto loadB = [&](int jt, v2f* bf) {
    const int j = jt * 16 + n;                // fused column 0..143
    const float* wbase; int stride, col;
    if (j < 64)       { wbase = W_in; stride = DINNER; col = j;       }
    else if (j < 128) { wbase = W_dt; stride = DINNER; col = j - 64;  }
    else if (j < 136) { wbase = W_B;  stride = DSTATE; col = j - 128; }
    else              { wbase = W_C;  stride = DSTATE; col = j - 136; }
    const float* wp = wbase + col + (2 * kg) * stride;
#pragma unroll
    for (int ks = 0; ks < 16; ++ks) {
      bf[ks].x = wp[(4 * ks)     * stride];   // Wcat[4ks+2kg  ][j]
      bf[ks].y = wp[(4 * ks + 1) * stride];   // Wcat[4ks+2kg+1][j]
    }
  };

  v2f bfA[16], bfB[16];
  loadB(0, bfA);

#pragma unroll 1
  for (int jt = 0; jt < 9; ++jt) {
    if (jt + 1 < 9) loadB(jt + 1, bfB);       // prefetch next tile's weights

    v8f acc = {0.f, 0.f, 0.f, 0.f, 0.f, 0.f, 0.f, 0.f};
#pragma unroll
    for (int ks = 0; ks < 16; ++ks)
      acc = __builtin_amdgcn_wmma_f32_16x16x4_f32(false, afrag[ks], false, bfA[ks],
                                                  (short)0, acc, false, false);

    const int j = jt * 16 + n;
    float bdt = 0.f;
    if (j >= 64 && j < 128) bdt = b_dt[j - 64];
#pragma unroll
    for (int r = 0; r < 8; ++r) {
      const int prow = p0 + r + 8 * kg;       // D row = r + 8*(lane>>4)
      const float v = acc[r];
      if (j < 64)       U [prow * DINNER + j]         = v;
      else if (j < 128) DT[prow * DINNER + (j - 64)]  = softplusf_(v + bdt);
      else if (j < 136) Bc[prow * DSTATE + (j - 128)] = v;
      else              Cc[prow * DSTATE + (j - 136)] = v;
    }
#pragma unroll
    for (int ks = 0; ks < 16; ++ks) bfA[ks] = bfB[ks];
  }
}

// ---------------------------------------------------------------------------
// K2: selective scan along W.  One wave per (d,h) sequence; each lane owns
// the contiguous channel pair e = 2*lane, 2*lane+1 so u/dt loads and the y
// store are single b64 accesses.  y overwrites U in place.
// ---------------------------------------------------------------------------
__global__ void __launch_bounds__(128)
scan_kernel(float* __restrict__ UY, const float* __restrict__ DT,
            const float* __restrict__ Bc, const float* __restrict__ Cc,
            const float* __restrict__ A_log, const float* __restrict__ D_skip)
{
  const int lane = threadIdx.x & 31;
  const int seq  = blockIdx.x * 4 + (threadIdx.x >> 5);  // = d*H + h, 0..1535
  const int e0 = 2 * lane;                               // channel pair

  float A0[DSTATE], A1[DSTATE];
#pragma unroll
  for (int s = 0; s < DSTATE; ++s) {
    A0[s] = -expf(A_log[e0 * DSTATE + s]);               // contiguous 16 floats
    A1[s] = -expf(A_log[(e0 + 1) * DSTATE + s]);
  }
  const float2 dsv = *(const float2*)(D_skip + e0);

  float h0[DSTATE] = {}, h1[DSTATE] = {};
  const int pbase = seq * WW;
  for (int w = 0; w < WW; ++w) {
    const int p = pbase + w;
    const float2 uv  = *(const float2*)(UY + p * DINNER + e0);
    const float2 dtv = *(const float2*)(DT + p * DINNER + e0);
    float Bv[8], Cv[8];
    *(float4*)&Bv[0] = *(const float4*)(Bc + p * DSTATE);
    *(float4*)&Bv[4] = *(const float4*)(Bc + p * DSTATE + 4);
    *(float4*)&Cv[0] = *(const float4*)(Cc + p * DSTATE);
    *(float4*)&Cv[4] = *(const float4*)(Cc + p * DSTATE + 4);
    if (w + 4 < WW) {                                    // cover load latency
      __builtin_prefetch(&UY[(p + 4) * DINNER + e0], 0, 3);
      __builtin_prefetch(&DT[(p + 4) * DINNER + e0], 0, 3);
    }
    const float du0 = dtv.x * uv.x, du1 = dtv.y * uv.y;
    float y0 = dsv.x * uv.x, y1 = dsv.y * uv.y;
#pragma unroll
    for (int s = 0; s < DSTATE; ++s) {
      h0[s] = h0[s] * expf(dtv.x * A0[s]) + du0 * Bv[s];
      h1[s] = h1[s] * expf(dtv.y * A1[s]) + du1 * Bv[s];
      y0 += h0[s] * Cv[s];
      y1 += h1[s] * Cv[s];
    }
    float2 yv; yv.x = y0; yv.y = y1;
    *(float2*)(UY + p * DINNER + e0) = yv;
  }
}

// ---------------------------------------------------------------------------
// K3: cost GEMM  [P x 64] * [64 x 8 (padded to 16)] -> COST[Dv,G,H,W].
// All A and B fragments preloaded, 16 back-to-back WMMAs.
// ---------------------------------------------------------------------------
__global__ void __launch_bounds__(128)
cost_gemm_kernel(const float* __restrict__ YS, const float* __restrict__ W_out,
                 float* __restrict__ COST)
{
  const int lane = threadIdx.x & 31;
  const int tile = blockIdx.x * 4 + (threadIdx.x >> 5);
  const int p0   = tile * 16;
  const int m    = lane & 15;
  const int kg   = lane >> 4;
  const int n    = m;

  v2f af[16], bf[16];
#pragma unroll
  for (int ks = 0; ks < 16; ++ks) {
    const int k = 4 * ks + 2 * kg;
    af[ks] = *(const v2f*)(YS + (p0 + m) * DINNER + k);  // contiguous K pair
    bf[ks].x = (n < GG) ? W_out[k * GG + n]       : 0.f;
    bf[ks].y = (n < GG) ? W_out[(k + 1) * GG + n] : 0.f;
  }

  v8f acc = {0.f, 0.f, 0.f, 0.f, 0.f, 0.f, 0.f, 0.f};
#pragma unroll
  for (int ks = 0; ks < 16; ++ks)
    acc = __builtin_amdgcn_wmma_f32_16x16x4_f32(false, af[ks], false, bf[ks],
                                                (short)0, acc, false, false);

  if (n < GG) {
#pragma unroll
    for (int r = 0; r < 8; ++r) {
      const int p = p0 + r + 8 * kg;
      const int w = p & (WW - 1);
      const int h = (p >> 7) & (HH - 1);
      const int d = p >> 13;
      COST[((d * GG + n) * HH + h) * WW + w] = acc[r];
    }
  }
}

// ---------------------------------------------------------------------------
// K4: channel attention.  One block per disparity: masked avg+max over (h,w)
// per group, shared MLP (8->4->8), sigmoid gate -> CH[d][g].
// ---------------------------------------------------------------------------
__global__ void __launch_bounds__(256)
chattn_kernel(const float* __restrict__ COST,
              const float* __restrict__ W1, const float* __restrict__ b1,
              const float* __restrict__ W2, const float* __restrict__ b2,
              float* __restrict__ CH)
{
  __shared__ float red[256];
  __shared__ float s_avg[GG], s_mx[GG];
  const int tid = threadIdx.x;
  const int d   = blockIdx.x;
  for (int g = 0; g < GG; ++g) {
    const float* cp = COST + (d * GG + g) * (HH * WW);
    float sum = 0.f, mx = -INFINITY;
    for (int idx = tid; idx < HH * WW; idx += 256) {
      const int w = idx & (WW - 1);
      if (w >= d) { const float v = cp[idx]; sum += v; mx = fmaxf(mx, v); }
    }
    red[tid] = sum; __syncthreads();
    for (int off = 128; off > 0; off >>= 1) {
      if (tid < off) red[tid] += red[tid + off];
      __syncthreads();
    }
    if (tid == 0) s_avg[g] = red[0] / (float)(HH * (WW - d));
    __syncthreads();
    red[tid] = mx; __syncthreads();
    for (int off = 128; off > 0; off >>= 1) {
      if (tid < off) red[tid] = fmaxf(red[tid], red[tid + off]);
      __syncthreads();
    }
    if (tid == 0) s_mx[g] = red[0];
    __syncthreads();
  }
  if (tid == 0) {
    float ha[4], hm[4];
    for (int k = 0; k < 4; ++k) {
      float za = b1[k], zm = b1[k];
      for (int g = 0; g < GG; ++g) {
        za += s_avg[g] * W1[g * 4 + k];
        zm += s_mx[g]  * W1[g * 4 + k];
      }
      ha[k] = fmaxf(za, 0.f); hm[k] = fmaxf(zm, 0.f);
    }
    for (int g = 0; g < GG; ++g) {
      float oa = b2[g], om = b2[g];
      for (int k = 0; k < 4; ++k) {
        oa += ha[k] * W2[k * GG + g];
        om += hm[k] * W2[k * GG + g];
      }
      CH[d * GG + g] = sigmoidf_(oa + om);
    }
  }
}

// ---------------------------------------------------------------------------
// K5: channel gate, spatial gate over g, mask, transpose to [1,G,Dv,H,W].
// ---------------------------------------------------------------------------
__global__ void __launch_bounds__(256)
final_kernel(const float* __restrict__ COST, const float* __restrict__ CH,
             const float* __restrict__ w_sp, const float* __restrict__ b_sp,
             float* __restrict__ out)
{
  const int p = blockIdx.x * 256 + threadIdx.x;    // exactly PTOT threads
  const int w = p & (WW - 1);
  const int h = (p >> 7) & (HH - 1);
  const int d = p >> 13;
  float c[GG];
  float av = 0.f, mx = -INFINITY;
#pragma unroll
  for (int g = 0; g < GG; ++g) {
    const float v = COST[((d * GG + g) * HH + h) * WW + w] * CH[d * GG + g];
    c[g] = v; av += v; mx = fmaxf(mx, v);
  }
  av *= (1.f / GG);
  const float sp   = sigmoidf_(w_sp[0] * av + w_sp[1] * mx + b_sp[0]);
  const float gate = (w >= d) ? sp : 0.f;
#pragma unroll
  for (int g = 0; g < GG; ++g)
    out[((g * DVv + d) * HH + h) * WW + w] = c[g] * gate;
}

// ---------------------------------------------------------------------------
extern "C" void kernel_launch(void* const* d_in, const int* in_sizes, int n_in,
                              void* d_out, int out_size, void* d_ws, size_t ws_size,
                              hipStream_t stream) {
  const float* featL  = (const float*)d_in[0];
  const float* featR  = (const float*)d_in[1];
  const float* W_in   = (const float*)d_in[2];
  const float* W_dt   = (const float*)d_in[3];
  const float* b_dt   = (const float*)d_in[4];
  const float* W_B    = (const float*)d_in[5];
  const float* W_C    = (const float*)d_in[6];
  const float* A_log  = (const float*)d_in[7];
  const float* D_skip = (const float*)d_in[8];
  const float* W_out  = (const float*)d_in[9];
  const float* W1     = (const float*)d_in[10];
  const float* b1     = (const float*)d_in[11];
  const float* W2     = (const float*)d_in[12];
  const float* b2     = (const float*)d_in[13];
  const float* w_sp   = (const float*)d_in[14];
  const float* b_sp   = (const float*)d_in[15];

  // workspace layout (~108 MB):
  //   U/YS : P*64 f32  (50 331 648 B)   -- y overwrites u in the scan
  //   DT   : P*64 f32  (50 331 648 B)
  //   Bc   : P*8  f32  ( 6 291 456 B)   -- reused as COST after the scan
  //   Cc   : P*8  f32  ( 6 291 456 B)
  //   CH   : 24*8 f32
  char* ws = (char*)d_ws;
  float* U    = (float*)(ws);
  float* DT   = (float*)(ws + 50331648ull);
  float* Bc   = (float*)(ws + 100663296ull);
  float* Cc   = (float*)(ws + 106954752ull);
  float* CH   = (float*)(ws + 113246208ull);
  float* YS   = U;    // in-place
  float* COST = Bc;   // Bc dead after scan

  proj_gemm_kernel<<<PTOT / 16 / 4, 128, 0, stream>>>(featL, featR, W_in, W_dt,
                                                      b_dt, W_B, W_C, U, DT, Bc, Cc);
  scan_kernel<<<(DVv * HH) / 4, 128, 0, stream>>>(U, DT, Bc, Cc, A_log, D_skip);
  cost_gemm_kernel<<<PTOT / 16 / 4, 128, 0, stream>>>(YS, W_out, COST);
  chattn_kernel<<<DVv, 256, 0, stream>>>(COST, W1, b1, W2, b2, CH);
  final_kernel<<<PTOT / 256, 256, 0, stream>>>(COST, CH, w_sp, b_sp, (float*)d_out);
}